// Channel_30872224924159
// MI455X (gfx1250) — compile-verified
//
#include <hip/hip_runtime.h>
#include <math.h>

// ---------------------------------------------------------------------------
// AWGN channel for MI455X (gfx1250, wave32).
//
// Bandwidth-bound: ~470 MB of traffic -> ~20us at 23.3 TB/s HBM.
// Strategy:
//   K1: per-block partial sums of x^2 (B128 loads, RT policy so x stays in the
//       192MB L2 for the second pass). Wave reduction via V_WMMA_F32_16X16X4_F32
//       with a ones B-matrix (1 WMMA + 7 VALU adds + 1 shfl = 32-lane sum).
//   K2: single wave reduces the 512 partials (same WMMA reduction), writes
//       stddev = sqrt(sum/(N*snr)) to scratch.
//   K3: elementwise AWGN map. x read RT (L2 hit), nr/ni read NT, all stores NT
//       (streamed once, never reused) so they don't evict x from L2.
// No atomics, no scratch zero-init: every ws slot is overwritten every call,
// so graph replay is deterministic.
// ---------------------------------------------------------------------------

typedef float v2f __attribute__((ext_vector_type(2)));
typedef float v4f __attribute__((ext_vector_type(4)));
typedef float v8f __attribute__((ext_vector_type(8)));

#define SNR_LIN   10.0f   // 10^(CHANNEL_SNR_DB/10), CHANNEL_SNR_DB = 10
#define NPART     512
#define RED_THR   256

// Sum `acc` across all 32 lanes of the wave using the fp32 WMMA unit.
// A-matrix (16x4 f32): lane m      holds {A[m][0], A[m][1]}   (m = 0..15)
//                      lane m+16   holds {A[m][2], A[m][3]}
// With A.v0 = acc, A.v1 = 0, B = ones(4x16), C = 0:
//   D[m][n] = rowsum(A[m]) = acc[m] + acc[m+16]   (independent of n)
// D layout: vgpr j, lanes 0-15 -> D[j][n]; lanes 16-31 -> D[j+8][n].
// So sum_j d[j] gives (r0..r7) on low lanes, (r8..r15) on high lanes;
// one shfl_xor(16) completes the 32-lane total (broadcast to every lane).
// EXEC must be all-ones at the call site (callers guarantee full blocks,
// reconverged control flow).
__device__ __forceinline__ float wave_sum_wmma_f32(float acc) {
    v2f a; a.x = acc;  a.y = 0.0f;
    v2f b; b.x = 1.0f; b.y = 1.0f;
    v8f c = {};
    c = __builtin_amdgcn_wmma_f32_16x16x4_f32(
            /*neg_a=*/false, a, /*neg_b=*/false, b,
            /*c_mod=*/(short)0, c, /*reuse_a=*/false, /*reuse_b=*/false);
    float s = ((c[0] + c[1]) + (c[2] + c[3])) + ((c[4] + c[5]) + (c[6] + c[7]));
    s += __shfl_xor(s, 16, 32);
    return s;
}

// K1: per-block partial sums of x^2. Launch with exactly NPART blocks of
// RED_THR threads; part[] is fully overwritten each call.
__global__ __launch_bounds__(RED_THR)
void power_partials_kernel(const float* __restrict__ x,
                           float* __restrict__ part,
                           long long nvec, long long tail_base, int ntail) {
    const v4f* __restrict__ x4 = (const v4f*)x;
    const long long stride = (long long)gridDim.x * blockDim.x;
    float acc = 0.0f;
    for (long long i = (long long)blockIdx.x * blockDim.x + threadIdx.x;
         i < nvec; i += stride) {
        v4f v = x4[i];   // regular-temporal: keep x resident in 192MB L2
        acc = fmaf(v.x, v.x, fmaf(v.y, v.y, fmaf(v.z, v.z, fmaf(v.w, v.w, acc))));
    }
    if (blockIdx.x == 0 && (int)threadIdx.x < ntail) {  // N%4 tail (none here)
        float t = x[tail_base + threadIdx.x];
        acc = fmaf(t, t, acc);
    }
    // converged here: EXEC all-ones for the WMMA reduction
    float wsum = wave_sum_wmma_f32(acc);

    __shared__ float lsum[RED_THR / 32];
    const int lane = threadIdx.x & 31;
    const int wid  = threadIdx.x >> 5;
    if (lane == 0) lsum[wid] = wsum;
    __syncthreads();
    if (threadIdx.x == 0) {
        float t = 0.0f;
#pragma unroll
        for (int k = 0; k < RED_THR / 32; ++k) t += lsum[k];
        part[blockIdx.x] = t;
    }
}

// K2: one wave folds NPART partials into stddev = sqrt(sum * inv_n_snr).
__global__ __launch_bounds__(32)
void finalize_stddev_kernel(const float* __restrict__ part,
                            float* __restrict__ sd, float inv_n_snr) {
    float acc = 0.0f;
#pragma unroll
    for (int k = threadIdx.x; k < NPART; k += 32) acc += part[k];
    float tot = wave_sum_wmma_f32(acc);   // full wave, EXEC all-ones
    if (threadIdx.x == 0) sd[0] = sqrtf(tot * inv_n_snr);
}

// K3: out_c[2j] = x + nr*s, out_c[2j+1] = ni*s (complex64 interleaved),
//     out_h[j] = 1. NT loads for nr/ni and NT stores everywhere so the
//     streamed 335MB doesn't evict x's L2-resident 67MB.
__global__ __launch_bounds__(256)
void awgn_kernel(const float* __restrict__ x,
                 const float* __restrict__ nr,
                 const float* __restrict__ ni,
                 const float* __restrict__ sd,
                 float* __restrict__ out_c,
                 float* __restrict__ out_h,
                 long long nvec, long long n) {
    const float s = sd[0];   // uniform scalar (cached)
    const v4f* __restrict__ x4  = (const v4f*)x;
    const v4f* __restrict__ nr4 = (const v4f*)nr;
    const v4f* __restrict__ ni4 = (const v4f*)ni;
    v4f* __restrict__ c4 = (v4f*)out_c;
    v4f* __restrict__ h4 = (v4f*)out_h;

    const long long stride = (long long)gridDim.x * blockDim.x;
    const long long t0 = (long long)blockIdx.x * blockDim.x + threadIdx.x;

    const v4f ones = {1.0f, 1.0f, 1.0f, 1.0f};
    for (long long i = t0; i < nvec; i += stride) {
        v4f xv = x4[i];                                  // RT: expect L2 hit
        v4f rv = __builtin_nontemporal_load(&nr4[i]);    // streamed once
        v4f iv = __builtin_nontemporal_load(&ni4[i]);    // streamed once
        v4f lo, hi;
        lo.x = fmaf(rv.x, s, xv.x);  lo.y = iv.x * s;
        lo.z = fmaf(rv.y, s, xv.y);  lo.w = iv.y * s;
        hi.x = fmaf(rv.z, s, xv.z);  hi.y = iv.z * s;
        hi.z = fmaf(rv.w, s, xv.w);  hi.w = iv.w * s;
        __builtin_nontemporal_store(lo,   &c4[2 * i]);
        __builtin_nontemporal_store(hi,   &c4[2 * i + 1]);
        __builtin_nontemporal_store(ones, &h4[i]);
    }
    // scalar tail for N % 4 != 0 (not taken for this shape)
    for (long long j = nvec * 4 + t0; j < n; j += stride) {
        float xv = x[j], rv = nr[j], iv = ni[j];
        out_c[2 * j]     = fmaf(rv, s, xv);
        out_c[2 * j + 1] = iv * s;
        out_h[j]         = 1.0f;
    }
}

extern "C" void kernel_launch(void* const* d_in, const int* in_sizes, int n_in,
                              void* d_out, int out_size, void* d_ws, size_t ws_size,
                              hipStream_t stream) {
    // setup_inputs order: channal_input, P (unused), noise_r, noise_i
    const float* x  = (const float*)d_in[0];
    const float* nr = (const float*)d_in[2];
    const float* ni = (const float*)d_in[3];

    const long long n    = (long long)in_sizes[0];   // 16,777,216
    const long long nvec = n >> 2;
    const int       ntail = (int)(n - (nvec << 2));

    float* part = (float*)d_ws;          // NPART partial sums
    float* sd   = part + NPART;          // 1 float: noise stddev
    float* out  = (float*)d_out;         // [2n complex interleaved][n ones]

    power_partials_kernel<<<NPART, RED_THR, 0, stream>>>(x, part, nvec,
                                                         nvec << 2, ntail);
    finalize_stddev_kernel<<<1, 32, 0, stream>>>(part, sd,
                                                 1.0f / ((float)n * SNR_LIN));
    awgn_kernel<<<2048, 256, 0, stream>>>(x, nr, ni, sd, out, out + 2 * n,
                                          nvec, n);
}